// Net_85581518340619
// MI455X (gfx1250) — compile-verified
//
#include <hip/hip_runtime.h>
#include <hip/hip_bf16.h>

typedef __attribute__((ext_vector_type(2))) float v2f;
typedef __attribute__((ext_vector_type(8))) float v8f;

#define EMBED  75
#define NSLICE 19          // ceil(75/4)
#define NEGS   5
#define WAVES_PER_BLOCK 8  // 256 threads / wave32

// select c[i] for i in 0..7 without dynamic VGPR indexing
__device__ __forceinline__ float vsel8(v8f c, int i) {
    float d = c[0];
    d = (i == 1) ? c[1] : d;
    d = (i == 2) ? c[2] : d;
    d = (i == 3) ? c[3] : d;
    d = (i == 4) ? c[4] : d;
    d = (i == 5) ? c[5] : d;
    d = (i == 6) ? c[6] : d;
    d = (i == 7) ? c[7] : d;
    return d;
}

// numerically-stable log(sigmoid(x)) = min(x,0) - log1p(exp(-|x|))
__device__ __forceinline__ float log_sigmoid(float x) {
    return fminf(x, 0.f) - log1pf(__expf(-fabsf(x)));
}

__global__ void zero_scalar_kernel(float* out) {
    if (threadIdx.x == 0 && blockIdx.x == 0) out[0] = 0.f;
}

__global__ void __launch_bounds__(256)
w2v_sgns_loss_kernel(const float* __restrict__ WI,
                     const float* __restrict__ WO,
                     const int*   __restrict__ x_idx,
                     const int*   __restrict__ y_idx,
                     const int*   __restrict__ neg_idx,
                     float* __restrict__ out,
                     int batch) {
    const int lane    = threadIdx.x & 31;
    const int wave    = threadIdx.x >> 5;
    const int tile    = blockIdx.x * WAVES_PER_BLOCK + wave;
    const int ntiles  = batch >> 4;     // 16 batch rows per wave tile

    __shared__ float red[WAVES_PER_BLOCK];

    float partial = 0.f;
    if (tile < ntiles) {
        const int b0   = tile << 4;
        const int row  = lane & 15;          // matrix row owned by this lane
        const int koff = (lane >> 4) << 1;   // 0 for lanes 0-15, 2 for lanes 16-31
        const int b    = b0 + row;

        // ---- A fragments: vI tile (16 x 75), kept resident in VGPRs ----
        const int xi = x_idx[b];
        const float* __restrict__ arow = WI + (long long)xi * EMBED;
        v2f afrag[NSLICE];
#pragma unroll
        for (int s = 0; s < NSLICE; ++s) {
            const int c0 = s * 4 + koff;           // <= 74 always
            afrag[s].x = arow[c0];
            afrag[s].y = (c0 + 1 < EMBED) ? arow[c0 + 1] : 0.f;  // pad K 75->76
        }

        float sum_pos = 0.f, sum_neg = 0.f;

#pragma unroll
        for (int p = 0; p < 1 + NEGS; ++p) {
            // partner row index: p==0 -> positive context, else negative sample
            const int pi = (p == 0) ? y_idx[b] : neg_idx[b * NEGS + (p - 1)];
            const float* __restrict__ brow = WO + (long long)pi * EMBED;

            // ---- B fragments: partner tile transposed (K x 16) ----
            v2f bfrag[NSLICE];
#pragma unroll
            for (int s = 0; s < NSLICE; ++s) {
                const int c0 = s * 4 + koff;
                bfrag[s].x = brow[c0];
                bfrag[s].y = (c0 + 1 < EMBED) ? brow[c0 + 1] : 0.f;
            }

            // ---- chained f32 WMMA over 19 K-slices: D = sum_s A_s x B_s ----
            v8f c = {0.f, 0.f, 0.f, 0.f, 0.f, 0.f, 0.f, 0.f};
#pragma unroll
            for (int s = 0; s < NSLICE; ++s) {
                c = __builtin_amdgcn_wmma_f32_16x16x4_f32(
                        /*neg_a=*/false, afrag[s],
                        /*neg_b=*/false, bfrag[s],
                        /*c_mod=*/(short)0, c,
                        /*reuse_a=*/false, /*reuse_b=*/false);
            }

            // ---- diagonal extraction: D[m][m] = dot(vI_m, partner_m) ----
            // lanes 0-7  hold diag m=lane    in c[lane]
            // lanes 24-31 hold diag m=lane-16 in c[lane-24]
            const float d     = vsel8(c, lane & 7);
            const bool  valid = (lane < 8) || (lane >= 24);
            const float z     = (p == 0) ? d : -d;   // neg_z = -dot
            float ls = log_sigmoid(z);
            ls = valid ? ls : 0.f;
            if (p == 0) sum_pos += ls; else sum_neg += ls;
        }
        partial = sum_pos * (1.f / (float)batch) + sum_neg;
    }

    // wave32 reduction
#pragma unroll
    for (int off = 16; off > 0; off >>= 1)
        partial += __shfl_xor(partial, off, 32);
    if (lane == 0) red[wave] = partial;
    __syncthreads();

    if (threadIdx.x == 0) {
        float s = 0.f;
#pragma unroll
        for (int i = 0; i < WAVES_PER_BLOCK; ++i) s += red[i];
        atomicAdd(out, -s);   // loss = -((1/B)*sum_pos + sum_neg)
    }
}

extern "C" void kernel_launch(void* const* d_in, const int* in_sizes, int n_in,
                              void* d_out, int out_size, void* d_ws, size_t ws_size,
                              hipStream_t stream) {
    const float* WI      = (const float*)d_in[0];
    const float* WO      = (const float*)d_in[1];
    const int*   x_idx   = (const int*)d_in[2];
    const int*   y_idx   = (const int*)d_in[3];
    const int*   neg_idx = (const int*)d_in[4];
    float*       out     = (float*)d_out;

    const int batch  = in_sizes[2];              // 262144
    const int ntiles = batch >> 4;               // 16384 wave tiles
    const int blocks = (ntiles + WAVES_PER_BLOCK - 1) / WAVES_PER_BLOCK;

    zero_scalar_kernel<<<1, 1, 0, stream>>>(out);
    w2v_sgns_loss_kernel<<<blocks, 32 * WAVES_PER_BLOCK, 0, stream>>>(
        WI, WO, x_idx, y_idx, neg_idx, out, batch);
}